// Memory_26645977104962
// MI455X (gfx1250) — compile-verified
//
#include <hip/hip_runtime.h>

#define N_NODES 100000
#define N_RELS  500
#define MEM_DIM 512
#define IN_DIM  1024
#define BATCH   65536

#define KCHUNK 64
#define LDSS   68      // KCHUNK + 4 pad: row r starts at bank (r*68)%64 = 4r -> all 64 banks used
#define MBLK   128     // batch rows per block (8 M-tiles)
#define NBLK   256     // output cols per block (16 N-tiles)

typedef __attribute__((ext_vector_type(2))) float v2f;
typedef __attribute__((ext_vector_type(8))) float v8f;

// ---------------------------------------------------------------------------
// Kernel 1: out = concat(entity_memory, rel_memory) * scale
// scale = time/(time+1) if time>1 else 1   (time read on device)
// ---------------------------------------------------------------------------
__global__ void __launch_bounds__(256)
mem_rescale_init_kernel(const float* __restrict__ ent,
                        const float* __restrict__ rel,
                        const int*   __restrict__ timep,
                        float*       __restrict__ out)
{
    const int n_ent4 = (N_NODES * MEM_DIM) / 4;
    const int n_tot4 = ((N_NODES + N_RELS) * MEM_DIM) / 4;
    int i = blockIdx.x * 256 + threadIdx.x;
    if (i >= n_tot4) return;

    int   tv    = *timep;
    float tf    = (float)tv;
    float scale = (tv > 1) ? (tf / (tf + 1.0f)) : 1.0f;

    float4 v = (i < n_ent4) ? reinterpret_cast<const float4*>(ent)[i]
                            : reinterpret_cast<const float4*>(rel)[i - n_ent4];
    v.x *= scale; v.y *= scale; v.z *= scale; v.w *= scale;
    reinterpret_cast<float4*>(out)[i] = v;
}

// ---------------------------------------------------------------------------
// Kernel 2: C = A @ W^T + b via f32 WMMA 16x16x4, then scatter-add
// (C * 1/(time+1)) into out[out_off + ids[row]*512 + col].
//
// Block: 256 threads = 8 wave32 = 2 M-groups x 4 N-groups.
// Block tile: 128 rows x 256 cols. Wave tile: 4 M-tiles x 4 N-tiles.
// A (128 x KCHUNK fp32) staged in LDS, W streamed from L2 (2 MB, resident).
// ---------------------------------------------------------------------------
__global__ void __launch_bounds__(256)
gemm_scatter_kernel(const float* __restrict__ A,      // [BATCH, IN_DIM]
                    const float* __restrict__ W,      // [MEM_DIM, IN_DIM]
                    const float* __restrict__ bias,   // [MEM_DIM]
                    const int*   __restrict__ ids,    // [BATCH]
                    const int*   __restrict__ timep,
                    float*       __restrict__ out,
                    unsigned long long out_off)
{
    __shared__ float As[MBLK * LDSS];   // 128*68*4 = 34816 B

    const int tid  = threadIdx.x;
    const int wave = tid >> 5;
    const int lane = tid & 31;
    const int half = lane >> 4;   // 0: K={0,1}, D rows 0-7 ; 1: K={2,3}, D rows 8-15
    const int nl   = lane & 15;   // M row of A frag == N col of B/D frag
    const int mg   = wave >> 2;   // M group: rows mg*64 .. mg*64+63 of block tile
    const int ng   = wave & 3;    // N group: cols ng*64 .. ng*64+63 of block tile

    const long long mbase = (long long)blockIdx.x * MBLK;
    const int       nbase = blockIdx.y * NBLK + ng * 64;

    v8f acc[4][4];   // [tm][tn]
#pragma unroll
    for (int tm = 0; tm < 4; ++tm)
#pragma unroll
        for (int tn = 0; tn < 4; ++tn)
#pragma unroll
            for (int r = 0; r < 8; ++r) acc[tm][tn][r] = 0.0f;

    for (int kc = 0; kc < IN_DIM; kc += KCHUNK) {
        __syncthreads();
        // Stage 128 x 64 fp32 A chunk: 2048 float4, 8 per thread.
        for (int i = tid; i < (MBLK * KCHUNK) / 4; i += 256) {
            int row = i >> 4;            // 16 float4 per row
            int c4  = (i & 15) << 2;
            float4 v = *reinterpret_cast<const float4*>(
                A + (mbase + row) * IN_DIM + kc + c4);
            *reinterpret_cast<float4*>(&As[row * LDSS + c4]) = v;
        }
        __syncthreads();

        for (int kk = 0; kk < KCHUNK; kk += 4) {
            const int ak = kk + half * 2;       // K pair for this half-wave
            v2f a[4], b[4];
#pragma unroll
            for (int tm = 0; tm < 4; ++tm) {
                const int arow = mg * 64 + tm * 16 + nl;
                float2 av = *reinterpret_cast<const float2*>(&As[arow * LDSS + ak]);
                a[tm].x = av.x; a[tm].y = av.y;
            }
#pragma unroll
            for (int tn = 0; tn < 4; ++tn) {
                const int n = nbase + tn * 16 + nl;   // W row = output column
                float2 bw = *reinterpret_cast<const float2*>(
                    W + (long long)n * IN_DIM + kc + ak);
                b[tn].x = bw.x; b[tn].y = bw.y;
            }
            // 16 WMMAs per k-step; 16 independent acc chains pipeline the XDL.
#pragma unroll
            for (int tm = 0; tm < 4; ++tm)
#pragma unroll
                for (int tn = 0; tn < 4; ++tn)
                    acc[tm][tn] = __builtin_amdgcn_wmma_f32_16x16x4_f32(
                        false, a[tm], false, b[tn], (short)0, acc[tm][tn],
                        false, false);
        }
    }

    const int   tv  = *timep;
    const float inv = 1.0f / ((float)tv + 1.0f);

    // D layout: lane = N (col nl), VGPR r -> M = r + half*8
#pragma unroll
    for (int tm = 0; tm < 4; ++tm) {
        int rid[8];
#pragma unroll
        for (int r = 0; r < 8; ++r)
            rid[r] = ids[mbase + mg * 64 + tm * 16 + half * 8 + r];
#pragma unroll
        for (int tn = 0; tn < 4; ++tn) {
            const int   ncol = nbase + tn * 16 + nl;
            const float bc   = bias[ncol];
#pragma unroll
            for (int r = 0; r < 8; ++r) {
                float val = (acc[tm][tn][r] + bc) * inv;
                unsafeAtomicAdd(out + out_off +
                                (unsigned long long)rid[r] * MEM_DIM + ncol, val);
            }
        }
    }
}

// ---------------------------------------------------------------------------
extern "C" void kernel_launch(void* const* d_in, const int* in_sizes, int n_in,
                              void* d_out, int out_size, void* d_ws, size_t ws_size,
                              hipStream_t stream)
{
    const float* nodes_emb = (const float*)d_in[0];
    const float* rels_emb  = (const float*)d_in[1];
    const int*   nodes_ids = (const int*)  d_in[2];
    const int*   rels_ids  = (const int*)  d_in[3];
    const float* ent_mem   = (const float*)d_in[4];
    const float* rel_mem   = (const float*)d_in[5];
    const float* W_node    = (const float*)d_in[6];
    const float* b_node    = (const float*)d_in[7];
    const float* W_rel     = (const float*)d_in[8];
    const float* b_rel     = (const float*)d_in[9];
    const int*   timep     = (const int*)  d_in[10];
    float*       out       = (float*)d_out;

    // 1) out = concat(entity_memory, rel_memory) * scale
    const int n_tot4 = ((N_NODES + N_RELS) * MEM_DIM) / 4;
    mem_rescale_init_kernel<<<(n_tot4 + 255) / 256, 256, 0, stream>>>(
        ent_mem, rel_mem, timep, out);

    // 2) out[ids] += (A @ W^T + b) / (time+1)   for nodes then rels
    dim3 grid(BATCH / MBLK, MEM_DIM / NBLK);   // (512, 2)
    gemm_scatter_kernel<<<grid, 256, 0, stream>>>(
        nodes_emb, W_node, b_node, nodes_ids, timep, out, 0ull);
    gemm_scatter_kernel<<<grid, 256, 0, stream>>>(
        rels_emb, W_rel, b_rel, rels_ids, timep, out,
        (unsigned long long)N_NODES * MEM_DIM);
}